// HashEncoder_49031346651385
// MI455X (gfx1250) — compile-verified
//
#include <hip/hip_runtime.h>

// Instant-NGP multiresolution hash encoding, CDNA5 (gfx1250) version.
//
// Design (see analysis): memory-bound gather kernel.
//  - one thread per (point, level):  t = n*16 + l
//  - output float2 index == t  -> 32 lanes store 256 contiguous bytes
//  - tables (64 MB) stay L2-resident; output (256 MB stream) bypasses
//    cache pressure via non-temporal stores (TH=NT)
//  - wave32 blocks of 256 threads

typedef float v2f __attribute__((ext_vector_type(2)));

#define HE_N_LEVELS     16u
#define HE_HASHMAP_SIZE (1u << 19)
#define HE_HASH_MASK    (HE_HASHMAP_SIZE - 1u)

__global__ __launch_bounds__(256) void HashEncoder_49031346651385_kernel(
    const float* __restrict__ x,       // [N, 3]
    const float* __restrict__ tables,  // [16, 2^19, 2]
    float*       __restrict__ out,     // [N, 32] level-major
    unsigned n_points)
{
    unsigned t = blockIdx.x * blockDim.x + threadIdx.x;   // (point, level) pair id
    unsigned n = t >> 4;          // point index
    unsigned l = t & 15u;         // level index
    if (n >= n_points) return;

    // --- load the point (3 floats; 16 lanes share them -> L0 hits) ---
    unsigned xb = 3u * n;
    float x0 = x[xb + 0u];
    float x1 = x[xb + 1u];
    float x2 = x[xb + 2u];

    // --- scale = 2^l via exponent bit trick (no transcendental) ---
    union { unsigned u; float f; } sc;
    sc.u = (l + 127u) << 23;

    // --- coords: floor(x * 2^l) as int32 -> uint32 (matches jnp path) ---
    unsigned c0 = (unsigned)(int)floorf(x0 * sc.f);
    unsigned c1 = (unsigned)(int)floorf(x1 * sc.f);
    unsigned c2 = (unsigned)(int)floorf(x2 * sc.f);

    // --- spatial hash with uint32 wraparound, mask to 19 bits ---
    unsigned h = (c0 + c1 * 2654435761u + c2 * 805459861u) & HE_HASH_MASK;

    // --- gather float2 feature from this level's table (L2-resident) ---
    const v2f* __restrict__ tab = (const v2f*)tables;
    unsigned idx = l * HE_HASHMAP_SIZE + h;   // < 2^23, fits 32-bit offset
    v2f feat = tab[idx];

    // --- streamed, never re-read output: non-temporal store to protect
    //     L2 residency of the hash tables. out float2 index == t. ---
    v2f* __restrict__ o = (v2f*)out;
    __builtin_nontemporal_store(feat, &o[t]);
}

extern "C" void kernel_launch(void* const* d_in, const int* in_sizes, int n_in,
                              void* d_out, int out_size, void* d_ws, size_t ws_size,
                              hipStream_t stream) {
    const float* x      = (const float*)d_in[0];  // [N,3] float32
    const float* tables = (const float*)d_in[1];  // [16, 2^19, 2] float32
    float*       out    = (float*)d_out;          // [N, 32] float32

    unsigned n_points = (unsigned)(in_sizes[0] / 3);
    unsigned total    = n_points * HE_N_LEVELS;   // one thread per (point, level)

    dim3 block(256);
    dim3 grid((total + block.x - 1u) / block.x);
    HashEncoder_49031346651385_kernel<<<grid, block, 0, stream>>>(
        x, tables, out, n_points);
}